// CP_34041910788864
// MI455X (gfx1250) — compile-verified
//
#include <hip/hip_runtime.h>
#include <math.h>

// ---------------------------------------------------------------------------
// VQ nearest/third codeword search for MI455X (gfx1250, wave32, WMMA).
// X = 13056 x 512 f32 rows, E = 2048 x 512 f32 codebook.
// Rank by d[i,j] = ||e_j||^2 - 2 * x_i . e_j   (||x_i||^2 constant per row).
// GEMM on V_WMMA_F32_16X16X4_F32 (exact f32). X tile staged to LDS with
// GLOBAL_LOAD_ASYNC_TO_LDS_B128 (+ s_wait_asynccnt). 32 rows/block so the two
// waves sharing each E column-quarter dedup codebook traffic in the WGP$.
// ---------------------------------------------------------------------------

typedef __attribute__((ext_vector_type(2))) float v2f;
typedef __attribute__((ext_vector_type(8))) float v8f;

#define EMB_DIM   512
#define NUM_EMB   2048
#define NROWS     13056        // 256*51
#define ROWS_BLK  32
#define ROW_BLKS  408          // NROWS / 32
#define MERGE_N   192          // 4 col-quarter waves * 16 lanes * 3 candidates

// ---------------- kernel 0: zero loss accumulators -------------------------
__global__ void vq_init_kernel(float* sums) {
    if (threadIdx.x < 2) sums[threadIdx.x] = 0.0f;
}

// ---------------- kernel 1: ||e_j||^2 --------------------------------------
__global__ __launch_bounds__(128)
void vq_emb_norm_kernel(const float* __restrict__ emb, float* __restrict__ enorm) {
    __shared__ float red[128];
    const int row = blockIdx.x;
    const float4 v = reinterpret_cast<const float4*>(emb + (size_t)row * EMB_DIM)[threadIdx.x];
    red[threadIdx.x] = v.x * v.x + v.y * v.y + v.z * v.z + v.w * v.w;
    __syncthreads();
    for (int off = 64; off > 0; off >>= 1) {
        if (threadIdx.x < off) red[threadIdx.x] += red[threadIdx.x + off];
        __syncthreads();
    }
    if (threadIdx.x == 0) enorm[row] = red[0];
}

// ---------------- kernel 2: WMMA distances + per-row top-3 -----------------
// Block = 256 threads (8 waves), owns 32 rows of X.
// wave w: row tile = w & 1 (16 rows), column quarter = w >> 1 (512 cols).
// X tile lives in a 64 KB LDS buffer, XOR-swizzled so the 16-lane A-fragment
// reads hit 16 distinct banks; the buffer is reused for the top-3 merge.
__global__ __launch_bounds__(256)
void vq_topk_kernel(const float* __restrict__ x, const float* __restrict__ emb,
                    const float* __restrict__ enorm,
                    int* __restrict__ nearest, int* __restrict__ third) {
    __shared__ __align__(16) unsigned char smem[ROWS_BLK * EMB_DIM * 4];  // 64 KB
    float* xs   = reinterpret_cast<float*>(smem);
    float* mval = reinterpret_cast<float*>(smem);                 // reused after barrier
    int*   midx = reinterpret_cast<int*>(smem + ROWS_BLK * MERGE_N * 4);

    const int tid     = threadIdx.x;
    const int wave    = tid >> 5;
    const int lane    = tid & 31;
    const int laneM   = lane & 15;
    const int halfSel = (lane < 16) ? 0 : 2;       // K sub-offset per lane half
    const int rowTile = wave & 1;                  // which 16-row tile
    const int colQ    = wave >> 1;                 // which 512-col quarter
    const int rowBase = blockIdx.x * ROWS_BLK;

    // ---- async-stage the 32x512 X tile into LDS (XOR bank swizzle) --------
    {
        const unsigned ldsBase = (unsigned)(uintptr_t)smem;   // low 32 bits = LDS offset
        for (int i = tid; i < ROWS_BLK * 128; i += 256) {     // 4096 float4's
            const int r  = i >> 7, c4 = i & 127;
            const int cf = (c4 * 4) ^ ((r & 15) * 4);         // swizzled float index
            const unsigned long long ga =
                (unsigned long long)(uintptr_t)(x + (size_t)(rowBase + r) * EMB_DIM + c4 * 4);
            const unsigned la = ldsBase + (unsigned)((r * EMB_DIM + cf) * 4);
            asm volatile("global_load_async_to_lds_b128 %0, %1, off"
                         :: "v"(la), "v"(ga) : "memory");
        }
        asm volatile("s_wait_asynccnt 0x0" ::: "memory");
    }
    __syncthreads();

    // A fragment: row = rowTile*16 + laneM, float col = (k + halfSel) ^ swz.
    const float* xrow = &xs[(rowTile * 16 + laneM) * EMB_DIM];
    const int    swz  = laneM * 4;                  // swizzle mask (bits >= 2 only)

    // Per-lane running top-3 for each of this lane's 8 C-matrix rows.
    float b0[8], b1[8], b2[8];
    int   i0[8], i1[8], i2[8];
#pragma unroll
    for (int r = 0; r < 8; ++r) {
        b0[r] = b1[r] = b2[r] = 3.0e38f;
        i0[r] = i1[r] = i2[r] = 0;
    }

    const int colWave = colQ * 512;
    for (int g = 0; g < 8; ++g) {                   // 8 groups of 4 col-tiles
        const int colBase = colWave + g * 64;

        v8f acc[4];
#pragma unroll
        for (int t = 0; t < 4; ++t)
#pragma unroll
            for (int r = 0; r < 8; ++r) acc[t][r] = 0.0f;

        // B fragment: E row = colBase + t*16 + laneM, K = k + halfSel.
        const float* eb0 = emb + (size_t)(colBase +  0 + laneM) * EMB_DIM + halfSel;
        const float* eb1 = emb + (size_t)(colBase + 16 + laneM) * EMB_DIM + halfSel;
        const float* eb2 = emb + (size_t)(colBase + 32 + laneM) * EMB_DIM + halfSel;
        const float* eb3 = emb + (size_t)(colBase + 48 + laneM) * EMB_DIM + halfSel;

        for (int k = 0; k < EMB_DIM; k += 4) {
            const int ka = ((k ^ swz) | halfSel);                    // swizzled A col
            const v2f a  = *reinterpret_cast<const v2f*>(&xrow[ka]); // ds_load_b64
            const v2f q0 = *reinterpret_cast<const v2f*>(eb0 + k);   // global b64
            const v2f q1 = *reinterpret_cast<const v2f*>(eb1 + k);
            const v2f q2 = *reinterpret_cast<const v2f*>(eb2 + k);
            const v2f q3 = *reinterpret_cast<const v2f*>(eb3 + k);
            acc[0] = __builtin_amdgcn_wmma_f32_16x16x4_f32(false, a, false, q0, (short)0, acc[0], false, false);
            acc[1] = __builtin_amdgcn_wmma_f32_16x16x4_f32(false, a, false, q1, (short)0, acc[1], false, false);
            acc[2] = __builtin_amdgcn_wmma_f32_16x16x4_f32(false, a, false, q2, (short)0, acc[2], false, false);
            acc[3] = __builtin_amdgcn_wmma_f32_16x16x4_f32(false, a, false, q3, (short)0, acc[3], false, false);
        }

        // Fold in ||e||^2 and update running top-3.
#pragma unroll
        for (int t = 0; t < 4; ++t) {
            const int col = colBase + t * 16 + laneM;
            const float en = enorm[col];
#pragma unroll
            for (int r = 0; r < 8; ++r) {
                const float v = en - 2.0f * acc[t][r];
                if (v < b0[r]) { b2[r] = b1[r]; i2[r] = i1[r];
                                 b1[r] = b0[r]; i1[r] = i0[r];
                                 b0[r] = v;     i0[r] = col; }
                else if (v < b1[r]) { b2[r] = b1[r]; i2[r] = i1[r];
                                      b1[r] = v;     i1[r] = col; }
                else if (v < b2[r]) { b2[r] = v;     i2[r] = col; }
            }
        }
    }

    __syncthreads();   // all waves done reading xs; safe to overlay merge arrays

    // Dump candidates: per row, 4 col-quarter waves x 16 lanes x 3 = 192.
    const int laneSlot = colQ * 16 + laneM;          // 0..63
#pragma unroll
    for (int r = 0; r < 8; ++r) {
        const int rowM = (lane < 16) ? r : (r + 8);  // C layout: rows r / r+8
        const int gr   = rowTile * 16 + rowM;        // row within block (0..31)
        float* mv = &mval[gr * MERGE_N + laneSlot * 3];
        int*   mi = &midx[gr * MERGE_N + laneSlot * 3];
        mv[0] = b0[r]; mv[1] = b1[r]; mv[2] = b2[r];
        mi[0] = i0[r]; mi[1] = i1[r]; mi[2] = i2[r];
    }
    __syncthreads();

    // One thread per row merges its 192 candidates.
    if (tid < ROWS_BLK) {
        float c0 = 3.0e38f, c1 = 3.0e38f, c2 = 3.0e38f;
        int   j0 = 0, j1 = 0, j2 = 0;
        const float* mv = &mval[tid * MERGE_N];
        const int*   mi = &midx[tid * MERGE_N];
        for (int s = 0; s < MERGE_N; ++s) {
            const float v = mv[s]; const int j = mi[s];
            if (v < c0)      { c2 = c1; j2 = j1; c1 = c0; j1 = j0; c0 = v; j0 = j; }
            else if (v < c1) { c2 = c1; j2 = j1; c1 = v;  j1 = j; }
            else if (v < c2) { c2 = v;  j2 = j; }
        }
        nearest[rowBase + tid] = j0;
        third[rowBase + tid]   = j2;
    }
}

// ---------------- kernel 3: gather + squared-error sums --------------------
__global__ __launch_bounds__(128)
void vq_gather_kernel(const float* __restrict__ x, const float* __restrict__ emb,
                      const int* __restrict__ nearest, const int* __restrict__ third,
                      float* __restrict__ out_fea, float* __restrict__ sums) {
    __shared__ float r1[128], r2[128];
    const int row = blockIdx.x;
    const int t   = threadIdx.x;
    const int jn  = nearest[row];
    const int j3  = third[row];

    const float4 xv = reinterpret_cast<const float4*>(x   + (size_t)row * EMB_DIM)[t];
    const float4 ev = reinterpret_cast<const float4*>(emb + (size_t)jn  * EMB_DIM)[t];
    const float4 tv = reinterpret_cast<const float4*>(emb + (size_t)j3  * EMB_DIM)[t];
    reinterpret_cast<float4*>(out_fea + (size_t)row * EMB_DIM)[t] = ev;

    const float d0 = ev.x - xv.x, d1 = ev.y - xv.y, d2 = ev.z - xv.z, d3 = ev.w - xv.w;
    const float e0 = tv.x - xv.x, e1 = tv.y - xv.y, e2 = tv.z - xv.z, e3 = tv.w - xv.w;
    r1[t] = d0 * d0 + d1 * d1 + d2 * d2 + d3 * d3;
    r2[t] = e0 * e0 + e1 * e1 + e2 * e2 + e3 * e3;
    __syncthreads();
    for (int off = 64; off > 0; off >>= 1) {
        if (t < off) { r1[t] += r1[t + off]; r2[t] += r2[t + off]; }
        __syncthreads();
    }
    if (t == 0) {
        atomicAdd(&sums[0], r1[0]);
        atomicAdd(&sums[1], r2[0]);
    }
}

// ---------------- kernel 4: finalize scalars -------------------------------
__global__ void vq_final_kernel(const float* __restrict__ sums, float* __restrict__ out) {
    const float s1 = sums[0], s2 = sums[1];
    out[0] = 1.0f - sqrtf(s1) / sqrtf(s2);                       // cp_score
    out[1] = 0.25f * s1 / ((float)NROWS * (float)EMB_DIM);       // k_loss
}

// ---------------------------------------------------------------------------
extern "C" void kernel_launch(void* const* d_in, const int* in_sizes, int n_in,
                              void* d_out, int out_size, void* d_ws, size_t ws_size,
                              hipStream_t stream) {
    const float* x   = (const float*)d_in[0];   // IP_score   (256*51*512)
    const float* emb = (const float*)d_in[1];   // emb_weight (2048*512)
    float* out = (float*)d_out;                 // [cp_score, k_loss, feature_EMA...]

    // Workspace layout:
    float* sums    = (float*)d_ws;                       // 2 floats
    float* enorm   = sums + 64;                          // 2048 floats
    int*   nearest = (int*)(enorm + NUM_EMB);            // 13056 ints
    int*   third   = nearest + NROWS;                    // 13056 ints

    vq_init_kernel<<<1, 32, 0, stream>>>(sums);
    vq_emb_norm_kernel<<<NUM_EMB, 128, 0, stream>>>(emb, enorm);
    vq_topk_kernel<<<ROW_BLKS, 256, 0, stream>>>(x, emb, enorm, nearest, third);
    vq_gather_kernel<<<NROWS, 128, 0, stream>>>(x, emb, nearest, third, out + 2, sums);
    vq_final_kernel<<<1, 1, 0, stream>>>(sums, out);
}